// V1Column_27831388078550
// MI455X (gfx1250) — compile-verified
//
#include <hip/hip_runtime.h>
#include <hip/hip_bf16.h>
#include <stdint.h>

#define NN    100000
#define NSYN  5
#define TT    20
#define EREC  2000000
#define NLGN  17400
#define ELGN  600000
#define NBKG  100
#define EBKG  200000
#define ETOT  (EREC + ELGN + EBKG)
#define DTC   1.0f

typedef __attribute__((ext_vector_type(2))) float v2f;
typedef __attribute__((ext_vector_type(8))) float v8f;

#if __has_builtin(__builtin_amdgcn_wmma_f32_16x16x4_f32)
#define HAVE_WMMA_F32 1
#endif

// ---- CDNA5 async global->LDS staging (gfx1250, tracked by ASYNCcnt) ----
__device__ __forceinline__ void async_lds_b64(void* lds, const void* g) {
    uint32_t l  = (uint32_t)(uintptr_t)lds;   // low 32 bits of flat LDS ptr == LDS byte address
    uint64_t ga = (uint64_t)(uintptr_t)g;
    asm volatile("global_load_async_to_lds_b64 %0, %1, off" :: "v"(l), "v"(ga) : "memory");
}
__device__ __forceinline__ void async_lds_b32(void* lds, const void* g) {
    uint32_t l  = (uint32_t)(uintptr_t)lds;
    uint64_t ga = (uint64_t)(uintptr_t)g;
    asm volatile("global_load_async_to_lds_b32 %0, %1, off" :: "v"(l), "v"(ga) : "memory");
}
__device__ __forceinline__ void async_wait0() {
    asm volatile("s_wait_asynccnt 0" ::: "memory");
}

__global__ void zero_kernel(float* __restrict__ p, int n) {
    int i = blockIdx.x * blockDim.x + threadIdx.x;
    if (i < n) p[i] = 0.0f;
}

// One thread per edge (rec | lgn | bkg concatenated). Stages (post,pre,w) through
// LDS via async-to-LDS, gathers the presynaptic spike, and only touches the
// 20B factor record + 5 float atomics when the spike fired (spikes are 0/1).
__global__ void edge_kernel(
    const float* __restrict__ lgn_spk, const float* __restrict__ bkg_spk,
    const int*   __restrict__ rec_idx, const float* __restrict__ rec_w, const float* __restrict__ rec_fac,
    const int*   __restrict__ lgn_idx, const float* __restrict__ lgn_w, const float* __restrict__ lgn_fac,
    const int*   __restrict__ bkg_idx, const float* __restrict__ bkg_w, const float* __restrict__ bkg_fac,
    const float* __restrict__ Z, float* __restrict__ inp, int t)
{
    __shared__ __align__(16) int   s_idx[2 * 256];
    __shared__ __align__(16) float s_w[256];
    const int e   = blockIdx.x * blockDim.x + threadIdx.x;
    const int tid = threadIdx.x;
    if (e >= ETOT) return;

    const int* ip; const float* wp; const float* fp; int eloc; int kind;
    if (e < EREC)              { kind = 0; eloc = e;               ip = rec_idx; wp = rec_w; fp = rec_fac; }
    else if (e < EREC + ELGN)  { kind = 1; eloc = e - EREC;        ip = lgn_idx; wp = lgn_w; fp = lgn_fac; }
    else                       { kind = 2; eloc = e - EREC - ELGN; ip = bkg_idx; wp = bkg_w; fp = bkg_fac; }

    // Stage edge record (post, pre, w) into LDS with async copies.
    async_lds_b64(&s_idx[2 * tid], &ip[2 * eloc]);
    async_lds_b32(&s_w[tid],       &wp[eloc]);
    __builtin_prefetch(&fp[5 * eloc], 0, 1);   // global_prefetch_b8: warm the factor line
    async_wait0();

    const int   post = s_idx[2 * tid];
    const int   pre  = s_idx[2 * tid + 1];
    const float wv   = s_w[tid];

    float spike;
    if (kind == 0) {
        // pre encodes pre_neuron + N*(delay-1); circular slot holding z_{t-1-k}
        int k    = pre / NN;
        int i    = pre - k * NN;
        int slot = (t + 9 - k) % 5;            // == ((t-1-k) mod 5), non-negative
        spike = Z[slot * NN + i];
    } else if (kind == 1) {
        spike = lgn_spk[t * NLGN + pre];       // B == 1
    } else {
        spike = bkg_spk[t * NBKG + pre];
    }

    if (spike != 0.0f) {                       // spikes are exactly 0.0 or 1.0
        float* o = inp + 5 * post;
        const float* f = fp + 5 * eloc;
        #pragma unroll
        for (int s = 0; s < NSYN; ++s) atomicAdd(o + s, wv * f[s]);
    }
}

// Element-wise GLIF neuron update. psc / psc_rise are SoA [s*NN + n] so the
// wave's psc block forms WMMA A-operands directly; the 4-synapse partial sum of
// input_current is computed on the FP32 matrix pipe (V_WMMA_F32_16X16X4_F32,
// B = all-ones so D[m][n] = sum_k A[m][k]), syn 4 added scalar.
// No early return: EXEC must be all-1s for WMMA; tail threads clamp loads and
// predicate stores.
__global__ void neuron_kernel(
    const float* __restrict__ decay, const float* __restrict__ current_factor,
    const float* __restrict__ v_reset, const float* __restrict__ t_ref,
    const float* __restrict__ kparam, const float* __restrict__ asc_amps,
    const float* __restrict__ v_th, const float* __restrict__ normalizer,
    const float* __restrict__ gathered_g,
    const float* __restrict__ syn_decay, const float* __restrict__ psc_initial,
    float* __restrict__ Z, float* __restrict__ v, float* __restrict__ r,
    float* __restrict__ asc, float* __restrict__ psc_rise, float* __restrict__ psc,
    float* __restrict__ inp, float* __restrict__ out, int t)
{
    const int  i    = blockIdx.x * blockDim.x + threadIdx.x;
    const bool live = (i < NN);
    const int  ic   = live ? i : (NN - 1);
    const int  lane = (int)(threadIdx.x & 31u);

    float sd[NSYN], pi[NSYN], pr[NSYN], pc[NSYN], in5[NSYN];
    #pragma unroll
    for (int s = 0; s < NSYN; ++s) {
        sd[s]  = syn_decay[s];
        pi[s]  = psc_initial[s];
        pr[s]  = psc_rise[s * NN + ic];
        pc[s]  = psc[s * NN + ic];        // OLD psc
        in5[s] = inp[ic * NSYN + s];
    }

    float input_current;
#ifdef HAVE_WMMA_F32
    {
        // Wave-uniform base neuron; rows of wmma#0 = nbase+0..15, wmma#1 = nbase+16..31.
        const int nbase = (int)(blockIdx.x * blockDim.x) + (int)(threadIdx.x & ~31u);
        const int lm    = lane & 15;
        const int sb    = (lane >= 16) ? 2 : 0;        // lanes 0-15: K=0,1 ; lanes 16-31: K=2,3
        int r0 = nbase + lm;        r0 = (r0 < NN) ? r0 : (NN - 1);
        int r1 = nbase + 16 + lm;   r1 = (r1 < NN) ? r1 : (NN - 1);
        v2f a0; a0.x = psc[sb * NN + r0]; a0.y = psc[(sb + 1) * NN + r0];
        v2f a1; a1.x = psc[sb * NN + r1]; a1.y = psc[(sb + 1) * NN + r1];
        v2f bones; bones.x = 1.0f; bones.y = 1.0f;     // all-ones B => row sums in D
        v8f cz = {};
        v8f d0 = __builtin_amdgcn_wmma_f32_16x16x4_f32(false, a0, false, bones,
                                                       (short)0, cz, false, false);
        v8f d1 = __builtin_amdgcn_wmma_f32_16x16x4_f32(false, a1, false, bones,
                                                       (short)0, cz, false, false);
        // D layout: lanes 0-15 hold rows 0-7 in VGPR0-7, lanes 16-31 hold rows 8-15.
        float e0 = d0[0], e1 = d1[0];
        #pragma unroll
        for (int j = 1; j < 8; ++j) {
            e0 = ((lane & 7) == j) ? d0[j] : e0;
            e1 = ((lane & 7) == j) ? d1[j] : e1;
        }
        // Route rowsum[m] to lane m: source lane = m (m<8) else m+8.
        const int src  = (lm < 8) ? lm : (lm + 8);
        const int bidx = src << 2;
        float s0 = __int_as_float(__builtin_amdgcn_ds_bpermute(bidx, __float_as_int(e0)));
        float s1 = __int_as_float(__builtin_amdgcn_ds_bpermute(bidx, __float_as_int(e1)));
        input_current = ((lane < 16) ? s0 : s1) + pc[4];
    }
#else
    input_current = pc[0] + pc[1] + pc[2] + pc[3] + pc[4];
#endif

    const float prev_z = Z[((t + 4) % 5) * NN + ic];  // z_{t-1} (slot 4 is zero at t=0)

    const float a0s = asc[ic * 2], a1s = asc[ic * 2 + 1];
    const float c1 = input_current + a0s + a1s + gathered_g[ic];
    const float dv = decay[ic] * v[ic] + current_factor[ic] * c1;
    const float newv = (prev_z > 0.5f) ? v_reset[ic] : dv;

    const float tr   = t_ref[ic];
    const float newr = fminf(fmaxf(r[ic] + prev_z * tr - DTC, 0.0f), tr);

    const float na0 = expf(-DTC * kparam[ic * 2])     * a0s + prev_z * asc_amps[ic * 2];
    const float na1 = expf(-DTC * kparam[ic * 2 + 1]) * a1s + prev_z * asc_amps[ic * 2 + 1];

    const float vsc  = (newv - v_th[ic]) / normalizer[ic];
    const float zv   = (vsc > 0.0f) ? 1.0f : 0.0f;
    const float newz = (newr > 0.0f) ? 0.0f : zv;

    if (live) {
        #pragma unroll
        for (int s = 0; s < NSYN; ++s) {
            psc_rise[s * NN + i] = pr[s] * sd[s] + in5[s] * pi[s];
            psc[s * NN + i]      = pc[s] * sd[s] + DTC * sd[s] * pr[s];  // OLD psc_rise
            inp[i * NSYN + s]    = 0.0f;   // ready for next step's edge kernel
        }
        v[i] = newv; r[i] = newr;
        asc[i * 2] = na0; asc[i * 2 + 1] = na1;
        Z[(t % 5) * NN + i] = newz;
        out[t * NN + i]     = newz;
    }
}

extern "C" void kernel_launch(void* const* d_in, const int* in_sizes, int n_in,
                              void* d_out, int out_size, void* d_ws, size_t ws_size,
                              hipStream_t stream) {
    const float* lgn_spikes     = (const float*)d_in[0];
    const float* bkg_spikes     = (const float*)d_in[1];
    const int*   rec_idx        = (const int*)  d_in[2];
    const float* rec_w          = (const float*)d_in[3];
    const float* rec_fac        = (const float*)d_in[4];
    const int*   lgn_idx        = (const int*)  d_in[5];
    const float* lgn_w          = (const float*)d_in[6];
    const float* lgn_fac        = (const float*)d_in[7];
    const int*   bkg_idx        = (const int*)  d_in[8];
    const float* bkg_w          = (const float*)d_in[9];
    const float* bkg_fac        = (const float*)d_in[10];
    const float* decay          = (const float*)d_in[11];
    const float* current_factor = (const float*)d_in[12];
    const float* v_reset        = (const float*)d_in[13];
    const float* t_ref          = (const float*)d_in[14];
    const float* kparam         = (const float*)d_in[15];
    const float* asc_amps       = (const float*)d_in[16];
    const float* v_th           = (const float*)d_in[17];
    const float* normalizer     = (const float*)d_in[18];
    const float* gathered_g     = (const float*)d_in[19];
    const float* syn_decay      = (const float*)d_in[20];
    const float* psc_initial    = (const float*)d_in[21];

    float* ws       = (float*)d_ws;
    float* Z        = ws;            // 5N  circular spike/delay buffer
    float* v        = ws + 5  * NN;  // N
    float* r        = ws + 6  * NN;  // N
    float* asc      = ws + 7  * NN;  // 2N
    float* psc_rise = ws + 9  * NN;  // 5N (SoA: [s*N + n])
    float* psc      = ws + 14 * NN;  // 5N (SoA: [s*N + n])
    float* inp      = ws + 19 * NN;  // 5N accumulation buffer (AoS: [n*5 + s])
    float* out      = (float*)d_out;

    const int state_elems = 24 * NN;
    zero_kernel<<<(state_elems + 255) / 256, 256, 0, stream>>>(ws, state_elems);

    const int eblocks = (ETOT + 255) / 256;
    const int nblocks = (NN + 255) / 256;
    for (int t = 0; t < TT; ++t) {
        edge_kernel<<<eblocks, 256, 0, stream>>>(
            lgn_spikes, bkg_spikes,
            rec_idx, rec_w, rec_fac,
            lgn_idx, lgn_w, lgn_fac,
            bkg_idx, bkg_w, bkg_fac,
            Z, inp, t);
        neuron_kernel<<<nblocks, 256, 0, stream>>>(
            decay, current_factor, v_reset, t_ref, kparam, asc_amps,
            v_th, normalizer, gathered_g, syn_decay, psc_initial,
            Z, v, r, asc, psc_rise, psc, inp, out, t);
    }
    (void)in_sizes; (void)n_in; (void)out_size; (void)ws_size;
}